// wokagconv_53867479826615
// MI455X (gfx1250) — compile-verified
//
#include <hip/hip_runtime.h>
#include <math.h>

#define B_     32
#define NHIS   12
#define GD     307
#define EMB    32
#define NPRE   12
#define HID    384      // EMB*NPRE
#define ROWS   (B_*GD)  // 9824

typedef __attribute__((ext_vector_type(2))) float v2f;
typedef __attribute__((ext_vector_type(8))) float v8f;

// ---------------------------------------------------------------------------
// Kernel 0: x3t[(b*307+g)*12 + t] = inputs[b, 1+t, 2+g]
// ---------------------------------------------------------------------------
__global__ __launch_bounds__(256) void transpose_x3_kernel(
    const float* __restrict__ inputs, float* __restrict__ x3t) {
  int idx = blockIdx.x * blockDim.x + threadIdx.x;
  if (idx >= ROWS * NHIS) return;
  int t = idx % NHIS;
  int r = idx / NHIS;
  int g = r % GD;
  int b = r / GD;
  x3t[idx] = inputs[(size_t)b * 13 * 309 + (size_t)(1 + t) * 309 + 2 + g];
}

// ---------------------------------------------------------------------------
// Kernel 1: per-batch reduction of dws over e.
//   Aw[b,t,i,j] = sum_e conv_w[e]*dws[date[b],t,e,i,j]
//   Cw[b,i,j]   = sum_{t,e} conv_b[e]*dws[date[b],t,e,i,j]
// grid = 32 blocks (one per batch), block = 384 threads (tid = i*12+j)
// ---------------------------------------------------------------------------
__global__ __launch_bounds__(HID) void precompute_kernel(
    const float* __restrict__ inputs, const float* __restrict__ conv_w,
    const float* __restrict__ conv_b, const float* __restrict__ dws,
    float* __restrict__ Aw, float* __restrict__ Cw) {
  int b = blockIdx.x;
  int tid = threadIdx.x;  // 0..383 == i*12+j
  int date = (int)inputs[(size_t)b * 13 * 309 + 1];
  const float* dwb = dws + (size_t)date * (NHIS * EMB * HID);
  float csum = 0.f;
  for (int t = 0; t < NHIS; ++t) {
    const float* dwt = dwb + (size_t)t * (EMB * HID);
    float asum = 0.f;
    for (int e = 0; e < EMB; ++e) {
      float w = dwt[(size_t)e * HID + tid];  // coalesced over tid
      asum += conv_w[e] * w;
      csum += conv_b[e] * w;
    }
    Aw[(size_t)(b * NHIS + t) * HID + tid] = asum;
  }
  Cw[(size_t)b * HID + tid] = csum;
}

// ---------------------------------------------------------------------------
// Kernel 2: x[b,g,i,j] = sum_t x3[b,t,g]*Aw[b,t,i,j] + Cw[b,i,j] + dbs[date,g,i,j]
// then LayerNorm over j (12) with gamma/beta, leaky-relu(0.3),
// stored as xe2[b,j,g,i]  (layout (B,12,307,32)).
// One lane per (b,g,i); 8 (b,g) groups of 32 lanes per 256-thread block.
// ---------------------------------------------------------------------------
__global__ __launch_bounds__(256) void xln_kernel(
    const float* __restrict__ inputs, const float* __restrict__ dbs,
    const float* __restrict__ Aw, const float* __restrict__ Cw,
    const float* __restrict__ ln_gamma, const float* __restrict__ ln_beta,
    float* __restrict__ xe2) {
  int p = blockIdx.x * 8 + (threadIdx.x >> 5);  // (b,g) pair, 0..9823
  int i = threadIdx.x & 31;                     // EMB index
  int b = p / GD;
  int g = p % GD;
  int date = (int)inputs[(size_t)b * 13 * 309 + 1];

  float x3v[NHIS];
  const float* x3b = inputs + (size_t)b * 13 * 309 + 309 + 2 + g;  // stride 309 in t
#pragma unroll
  for (int t = 0; t < NHIS; ++t) x3v[t] = x3b[(size_t)t * 309];

  const float* db = dbs + (size_t)date * (GD * HID) + (size_t)g * HID + i * NPRE;
  const float* cw = Cw + (size_t)b * HID + i * NPRE;

  float val[NPRE];
  float s = 0.f;
#pragma unroll
  for (int j = 0; j < NPRE; ++j) {
    float acc = cw[j] + db[j];
#pragma unroll
    for (int t = 0; t < NHIS; ++t)
      acc += x3v[t] * Aw[(size_t)(b * NHIS + t) * HID + i * NPRE + j];
    val[j] = acc;
    s += acc;
  }
  float mean = s * (1.f / 12.f);
  float var = 0.f;
#pragma unroll
  for (int j = 0; j < NPRE; ++j) {
    float d = val[j] - mean;
    var += d * d;
  }
  var *= (1.f / 12.f);
  float rs = rsqrtf(var + 0.001f);
#pragma unroll
  for (int j = 0; j < NPRE; ++j) {
    float y = (val[j] - mean) * rs * ln_gamma[j] + ln_beta[j];
    y = (y >= 0.f) ? y : 0.3f * y;
    // xe2[b][j][g][i]
    xe2[(size_t)b * (NHIS * GD * EMB) + (size_t)j * (GD * EMB) + (size_t)g * EMB + i] = y;
  }
}

// ---------------------------------------------------------------------------
// Generic fp32 WMMA GEMM:  C[M,N] = act(A[M,K] @ W[K,N] + bias)
// A row stride = K, W row-major K x N, C row-major M x N.
// M % 32 == 0, K % 4 == 0; N arbitrary (clamp+mask, no divergent branches).
// block = 256 threads (8 waves); macro-tile 32(M) x 64(N):
//   wave w -> rows (w>>2)*16, cols (w&3)*16.  Main K-loop over K&~31 has zero
//   guards; tail chunk (K=12 case) is branchless clamp+mask.  Inner loop is a
//   fully unrolled, unconditional run of 8 V_WMMA_F32_16X16X4_F32 fed from LDS.
// LDS layouts:
//   Ash: A[m][k] at m*36 + k          (stride 36 -> conflict-free wmma reads)
//   Bsh: W[k][c] at (k>>1)*160 + 2c + (k&1)
//        (pair-interleave -> {W[k][n],W[k+1][n]} is one aligned ds_load_b64;
//         pair-stride 160 puts lane halves on disjoint bank halves)
// ---------------------------------------------------------------------------
#define KCHUNK 32
__global__ __launch_bounds__(256) void gemm_wmma_kernel(
    const float* __restrict__ A, const float* __restrict__ W,
    const float* __restrict__ bias, float* __restrict__ C,
    int M, int N, int K, int act) {
  __shared__ float Ash[32 * 36];
  __shared__ float Bsh[16 * 160];

  const int tid = threadIdx.x;
  const int lane = tid & 31;
  const int wave = tid >> 5;
  const int wn = wave & 3;            // column tile within block
  const int wm = wave >> 2;           // row half within block
  const int lmod = lane & 15;
  const int lhalf = lane >> 4;        // 0 or 1
  const int koff = lhalf * 2;         // ISA: A/B vgpr pair holds K={0,1}/{2,3}
  const int mbase = blockIdx.y * 32;
  const int nblk = blockIdx.x * 64;
  const int cn = wn * 16 + lmod;      // column within 64-wide tile
  const int n = nblk + cn;
  const int arowL = wm * 16 + lmod;   // this wave's A row in LDS

  v8f acc = {0.f, 0.f, 0.f, 0.f, 0.f, 0.f, 0.f, 0.f};

  // staging thread mapping
  const int arow = tid >> 3;          // 0..31
  const int akk = (tid & 7) * 4;      // 0,4,...,28
  const int brow0 = tid >> 4;         // 0..15
  const int bcol = (tid & 15) * 4;    // 0,4,...,60

  // branchless N-tail handling for B staging: clamped offsets + 0/1 masks
  int coff[4];
  float cmask[4];
#pragma unroll
  for (int u = 0; u < 4; ++u) {
    int c = nblk + bcol + u;
    bool ok = (c < N);
    coff[u] = ok ? c : 0;
    cmask[u] = ok ? 1.f : 0.f;
  }

  const float* aprow = A + (size_t)(mbase + arow) * K;

  auto mma_chunk = [&]() {
#pragma unroll
    for (int kk = 0; kk < KCHUNK; kk += 4) {
      v2f a = *(const v2f*)&Ash[arowL * 36 + kk + koff];
      v2f bb = *(const v2f*)&Bsh[((kk + koff) >> 1) * 160 + cn * 2];
      acc = __builtin_amdgcn_wmma_f32_16x16x4_f32(
          /*neg_a=*/false, a, /*neg_b=*/false, bb,
          /*c_mod=*/(short)0, acc, /*reuse_a=*/false, /*reuse_b=*/false);
    }
  };

  const int Kfull = K & ~(KCHUNK - 1);
  for (int k0 = 0; k0 < Kfull; k0 += KCHUNK) {
    // ---- stage A tile (32 x 32): one unconditional float4 per thread ----
    *(float4*)&Ash[arow * 36 + akk] = *(const float4*)&aprow[k0 + akk];
    // ---- stage B tile (32 x 64): unconditional loads, mask-multiplied ----
#pragma unroll
    for (int it = 0; it < 2; ++it) {
      int r = brow0 + it * 16;        // k within chunk
      const float* wrow = W + (size_t)(k0 + r) * N;
      float* dst = &Bsh[(r >> 1) * 160 + bcol * 2 + (r & 1)];
#pragma unroll
      for (int u = 0; u < 4; ++u) dst[u * 2] = wrow[coff[u]] * cmask[u];
    }
    __syncthreads();
    mma_chunk();
    __syncthreads();
  }

  if (Kfull < K) {  // K-tail chunk (K % 32 != 0, e.g. K == 12): clamp+mask
#pragma unroll
    for (int u = 0; u < 4; ++u) {
      int k = Kfull + akk + u;
      bool ok = (k < K);
      Ash[arow * 36 + akk + u] = aprow[ok ? k : 0] * (ok ? 1.f : 0.f);
    }
#pragma unroll
    for (int it = 0; it < 2; ++it) {
      int r = brow0 + it * 16;
      int kr = Kfull + r;
      bool kok = (kr < K);
      const float* wrow = W + (size_t)(kok ? kr : 0) * N;
      float km = kok ? 1.f : 0.f;
      float* dst = &Bsh[(r >> 1) * 160 + bcol * 2 + (r & 1)];
#pragma unroll
      for (int u = 0; u < 4; ++u) dst[u * 2] = wrow[coff[u]] * (cmask[u] * km);
    }
    __syncthreads();
    mma_chunk();
  }

  if (n < N) {
    float bv = bias ? bias[n] : 0.f;
#pragma unroll
    for (int r = 0; r < 8; ++r) {
      int m = mbase + wm * 16 + r + lhalf * 8;  // lanes 16-31 hold rows M+8
      float v = acc[r] + bv;
      if (act == 1) v = (v > 0.f) ? v : 0.f;
      C[(size_t)m * N + n] = v;
    }
  }
}

// ---------------------------------------------------------------------------
// Kernel 4: gating + fully_w + denormalize.
//  g1[t] = (tg[b,t,g] + x1[b,g,t]) * sigmoid(tg[b,t,g])
//  g2[t] = (gt[b,g,t] + x2[b,g,t]) * sigmoid(gt[b,g,t])
//  out[b,o,g] = (sum_t g1[t]*Fw[t,o] + g2[t]*Fw[12+t,o]) * 50 + 60
// ---------------------------------------------------------------------------
__global__ __launch_bounds__(256) void final_kernel(
    const float* __restrict__ tg, const float* __restrict__ gt,
    const float* __restrict__ x1, const float* __restrict__ x2,
    const float* __restrict__ fully_w, float* __restrict__ out) {
  int p = blockIdx.x * blockDim.x + threadIdx.x;  // (b,g)
  if (p >= ROWS) return;
  int b = p / GD;
  int g = p % GD;
  float g1[NHIS], g2[NHIS];
#pragma unroll
  for (int t = 0; t < NHIS; ++t) {
    float tgv = tg[(size_t)(b * NHIS + t) * GD + g];
    float s1 = 1.f / (1.f + __expf(-tgv));
    g1[t] = (tgv + x1[(size_t)p * NHIS + t]) * s1;
    float gtv = gt[(size_t)p * NHIS + t];
    float s2 = 1.f / (1.f + __expf(-gtv));
    g2[t] = (gtv + x2[(size_t)p * NHIS + t]) * s2;
  }
#pragma unroll
  for (int o = 0; o < NPRE; ++o) {
    float s = 0.f;
#pragma unroll
    for (int t = 0; t < NHIS; ++t)
      s += g1[t] * fully_w[t * NPRE + o] + g2[t] * fully_w[(NHIS + t) * NPRE + o];
    out[(size_t)b * (NPRE * GD) + (size_t)o * GD + g] = s * 50.f + 60.f;
  }
}

// ---------------------------------------------------------------------------
static inline void launch_gemm(const float* A, const float* W, const float* bias,
                               float* C, int M, int N, int K, int act,
                               hipStream_t s) {
  dim3 grid((N + 63) / 64, M / 32);
  gemm_wmma_kernel<<<grid, 256, 0, s>>>(A, W, bias, C, M, N, K, act);
}

extern "C" void kernel_launch(void* const* d_in, const int* in_sizes, int n_in,
                              void* d_out, int out_size, void* d_ws, size_t ws_size,
                              hipStream_t stream) {
  const float* inputs   = (const float*)d_in[0];
  const float* conv_w   = (const float*)d_in[1];
  const float* conv_b   = (const float*)d_in[2];
  const float* dws      = (const float*)d_in[3];
  const float* dbs      = (const float*)d_in[4];
  const float* ln_gamma = (const float*)d_in[5];
  const float* ln_beta  = (const float*)d_in[6];
  const float* tge2tg   = (const float*)d_in[7];
  const float* gte2gt   = (const float*)d_in[8];
  const float* fully_w  = (const float*)d_in[9];
  const float* se_p[8]  = {(const float*)d_in[10], (const float*)d_in[11],
                           (const float*)d_in[12], (const float*)d_in[13],
                           (const float*)d_in[14], (const float*)d_in[15],
                           (const float*)d_in[16], (const float*)d_in[17]};
  const float* te_p[8]  = {(const float*)d_in[18], (const float*)d_in[19],
                           (const float*)d_in[20], (const float*)d_in[21],
                           (const float*)d_in[22], (const float*)d_in[23],
                           (const float*)d_in[24], (const float*)d_in[25]};

  float* ws = (float*)d_ws;
  float* x3t = ws;                       // 9824*12        = 117888
  float* Aw  = x3t + 117888;             // 32*12*384      = 147456
  float* Cw  = Aw + 147456;              // 32*384         = 12288
  float* xe2 = Cw + 12288;               // 32*12*307*32   = 3772416
  float* h_a = xe2 + 3772416;            // 9824*384       = 3772416
  float* h_b = h_a + 3772416;            // 9824*384       = 3772416
  float* x1  = h_b + 3772416;            // 9824*12        = 117888
  float* x2  = x1 + 117888;              // 117888
  float* tg  = x2 + 117888;              // 384*307        = 117888
  float* gt  = tg + 117888;              // 9824*12        = 117888

  // Stage 0: x3 transpose
  transpose_x3_kernel<<<(ROWS * NHIS + 255) / 256, 256, 0, stream>>>(inputs, x3t);
  // Stage 1: per-batch dws reduction over e
  precompute_kernel<<<B_, HID, 0, stream>>>(inputs, conv_w, conv_b, dws, Aw, Cw);
  // Stage 2: x + LayerNorm + leaky-relu -> xe2 layout (B,12,307,32)
  xln_kernel<<<ROWS / 8, 256, 0, stream>>>(inputs, dbs, Aw, Cw, ln_gamma, ln_beta, xe2);

  // Stage 3: SE MLP (x3t -> x1)
  launch_gemm(x3t, se_p[0], se_p[1], h_a, ROWS, HID, NHIS, 1, stream);
  launch_gemm(h_a, se_p[2], se_p[3], h_b, ROWS, HID, HID, 0, stream);
  launch_gemm(h_b, se_p[4], se_p[5], h_a, ROWS, HID, HID, 1, stream);
  launch_gemm(h_a, se_p[6], se_p[7], x1, ROWS, NPRE, HID, 0, stream);
  // Stage 4: TE MLP (x3t -> x2)
  launch_gemm(x3t, te_p[0], te_p[1], h_a, ROWS, HID, NHIS, 1, stream);
  launch_gemm(h_a, te_p[2], te_p[3], h_b, ROWS, HID, HID, 0, stream);
  launch_gemm(h_b, te_p[4], te_p[5], h_a, ROWS, HID, HID, 1, stream);
  launch_gemm(h_a, te_p[6], te_p[7], x2, ROWS, NPRE, HID, 0, stream);

  // Stage 5: tg[(b,t),v] = xe2(384 x 9824) @ tge2tg(9824 x 307)
  launch_gemm(xe2, tge2tg, nullptr, tg, B_ * NHIS, GD, GD * EMB, 0, stream);
  // Stage 6: gt[(b,g),o] = xe2-as-(9824 x 384) @ gte2gt(384 x 12)
  launch_gemm(xe2, gte2gt, nullptr, gt, ROWS, NPRE, NHIS * EMB, 0, stream);

  // Stage 7: gating + fully connected + denorm
  final_kernel<<<(ROWS + 255) / 256, 256, 0, stream>>>(tg, gt, x1, x2, fully_w,
                                                       (float*)d_out);
}